// Temporal_Decomposition_DWT_GPU_40776419508527
// MI455X (gfx1250) — compile-verified
//
#include <hip/hip_runtime.h>

// ---------------------------------------------------------------------------
// 2-level db4 wavelet low/high decomposition, MI455X (gfx1250, wave32).
//   xl = waverec([cA2,0,0]),  xh = x - xl   (exact by linearity of idwt)
// x tile staged via GLOBAL_LOAD_ASYNC_TO_LDS_B128 (ASYNCcnt path);
// analysis (symmetric ext) in VALU/LDS; 2-level synthesis folded into one
// banded 16x12 fp32 matrix applied per 16x16 tile via V_WMMA_F32_16X16X4_F32.
// ---------------------------------------------------------------------------

typedef __attribute__((ext_vector_type(2))) float v2f;
typedef __attribute__((ext_vector_type(8))) float v8f;

// db4 reconstruction low-pass (== REC_LO == DEC_LO reversed)
constexpr double H_D[8] = {
    0.23037781330885523,  0.7148465705525415,  0.6308807679295904,
   -0.02798376941698385, -0.18703481171888114, 0.030841381835986965,
    0.032883011666982945,-0.010597401784997278};

#define T_LEN   1024
#define N_LEN   512
#define NT      16      // n-tile width (one WMMA N dimension)
#define PITCH   20      // LDS row pitch (floats): 16B-aligned rows, no bank conflicts
#define CA1_LEN 515     // (1024+13-8)/2+1
#define CA2_LEN 261     // (515+13-8)/2+1
#define A2_ROWS 264     // cA2 rows + zero pad so WMMA B loads are unconditional

// Composite 2-level synthesis operator, representative tile t0=16 (period-4
// shift invariant; window j = t0/4 + jj, jj in [0,12)):
//   xl[t] = sum_k REC_LO[k] * al[(t+6-k)/2]    (t+6-k even, al trimmed to 515)
//   al[s] = sum_k2 REC_LO[k2] * cA2[(s+6-k2)/2]  (s+6-k2 even)
// Interior s-range never clips; only the cA2 j-range clips (handled by the
// zero-padded rows 261..263), so ONE matrix serves every tile exactly.
struct UTab {
  float u[16][12];
  constexpr UTab() : u{} {
    for (int i = 0; i < 16; ++i) {
      const int t = 16 + i;                  // representative tile, jw = 4
      for (int k = 0; k < 8; ++k) {
        const int v = t + 6 - k;
        if (v & 1) continue;
        const int s = v / 2;
        for (int k2 = 0; k2 < 8; ++k2) {
          const int w2 = s + 6 - k2;
          if (w2 & 1) continue;
          const int jj = w2 / 2 - 4;         // j - t0/4
          if (jj >= 0 && jj < 12) u[i][jj] += (float)(H_D[k] * H_D[k2]);
        }
      }
    }
  }
};
__device__ const UTab U_TAB{};

__global__ __launch_bounds__(256)
void dwt2_lowhigh_kernel(const float* __restrict__ x,
                         float* __restrict__ xl,
                         float* __restrict__ xh) {
  extern __shared__ float smem[];
  float* xs = smem;                          // [1024][PITCH] staged x tile
  float* a1 = xs + T_LEN * PITCH;            // [520][PITCH]  cA1
  float* a2 = a1 + 520 * PITCH;              // [264][PITCH]  cA2 (+zero pad)

  const int tid = threadIdx.x;
  const int nn  = tid & 15;                  // column within n-tile
  const int tt  = tid >> 4;                  // 0..15 row-group
  const int n0  = blockIdx.x * NT;           // 0..511 step 16
  const int b   = blockIdx.y;                // 0..63
  const float* xb = x + (size_t)b * T_LEN * N_LEN;

  float h[8];
  #pragma unroll
  for (int m = 0; m < 8; ++m) h[m] = (float)H_D[m];

  // ---- Stage 0: async-stage x tile straight into LDS (no VGPR round-trip).
  // Each lane moves 16B: row t = tid/4, float4-group c = tid%4.
  {
    const int c4 = tid & 3;
    const int tr = tid >> 2;                 // 0..63
    const unsigned ldsbase = (unsigned)(size_t)xs;   // LDS byte offset (flat lo32)
    #pragma unroll 4
    for (int s = 0; s < 16; ++s) {
      const int t = tr + 64 * s;
      const unsigned ldsa = ldsbase + (unsigned)(t * PITCH + 4 * c4) * 4u;
      const unsigned voff = (unsigned)(t * N_LEN + n0 + 4 * c4) * 4u;
      asm volatile("global_load_async_to_lds_b128 %0, %1, %2"
                   :: "v"(ldsa), "v"(voff), "s"(xb) : "memory");
    }
    asm volatile("s_wait_asynccnt 0x0" ::: "memory");
  }
  __syncthreads();

  // ---- Stage 1: cA1[i] = sum_m H[m] * x_sym(2i-6+m),  i in [0,515) ----
  for (int s = 0; s <= 32; ++s) {
    const int i = tt + 16 * s;
    if (i < CA1_LEN) {
      float acc = 0.f;
      #pragma unroll
      for (int m = 0; m < 8; ++m) {
        int q = 2 * i - 6 + m;
        q = (q < 0) ? (-q - 1) : q;                    // symmetric reflect
        q = (q >= T_LEN) ? (2 * T_LEN - 1 - q) : q;
        acc = fmaf(h[m], xs[q * PITCH + nn], acc);
      }
      a1[i * PITCH + nn] = acc;
    }
  }
  __syncthreads();

  // ---- Stage 2: cA2[i] = sum_m H[m] * cA1_sym(2i-6+m); zero-pad 261..263 ----
  for (int s = 0; s <= 16; ++s) {
    const int i = tt + 16 * s;
    if (i < CA2_LEN) {
      float acc = 0.f;
      #pragma unroll
      for (int m = 0; m < 8; ++m) {
        int q = 2 * i - 6 + m;
        q = (q < 0) ? (-q - 1) : q;
        q = (q >= CA1_LEN) ? (2 * CA1_LEN - 1 - q) : q;
        acc = fmaf(h[m], a1[q * PITCH + nn], acc);
      }
      a2[i * PITCH + nn] = acc;
    } else if (i < A2_ROWS) {
      a2[i * PITCH + nn] = 0.f;              // makes B-fragment loads unconditional
    }
  }
  __syncthreads();

  // ---- Stage 3: xl(16x16 tile) = U(16x12) x cA2window(12x16) via WMMA ----
  const int wave = tid >> 5;                 // 0..7
  const int lane = tid & 31;
  const int half = lane >> 4;                // K-pair select per ISA layout
  const int lm   = lane & 15;                // M (A,C/D) or N (B) index

  // A fragments are tile-invariant: lane<16 -> K={0,1}, lane>=16 -> K={2,3}
  v2f afr[3];
  #pragma unroll
  for (int kb = 0; kb < 3; ++kb) {
    afr[kb].x = U_TAB.u[lm][4 * kb + 2 * half + 0];
    afr[kb].y = U_TAB.u[lm][4 * kb + 2 * half + 1];
  }

  for (int s = 0; s < 8; ++s) {
    const int tile = wave + 8 * s;           // 0..63
    const int t0 = tile * 16;
    const int jw = tile * 4;                 // cA2 window start (= t0/4)
    v8f c = {};
    #pragma unroll
    for (int kb = 0; kb < 3; ++kb) {
      const int j0 = jw + 4 * kb + 2 * half; // <= 263, inside padded buffer
      v2f bfr;
      bfr.x = a2[ j0      * PITCH + lm];
      bfr.y = a2[(j0 + 1) * PITCH + lm];
      c = __builtin_amdgcn_wmma_f32_16x16x4_f32(
              /*neg_a=*/false, afr[kb], /*neg_b=*/false, bfr,
              /*c_mod=*/(short)0, c, /*reuse_a=*/false, /*reuse_b=*/false);
    }
    // C/D layout: VGPR v -> row M = v + 8*half, col N = lm.
    // Preload the 8 x values so DS loads pipeline instead of serializing.
    float xv[8];
    #pragma unroll
    for (int v = 0; v < 8; ++v)
      xv[v] = xs[(t0 + v + 8 * half) * PITCH + lm];

    const size_t outbase = ((size_t)b * T_LEN + t0) * N_LEN + n0 + lm;
    #pragma unroll
    for (int v = 0; v < 8; ++v)
      xl[outbase + (size_t)(v + 8 * half) * N_LEN] = c[v];
    #pragma unroll
    for (int v = 0; v < 8; ++v)
      xh[outbase + (size_t)(v + 8 * half) * N_LEN] = xv[v] - c[v];  // xh = x - xl
  }
}

extern "C" void kernel_launch(void* const* d_in, const int* in_sizes, int n_in,
                              void* d_out, int out_size, void* d_ws, size_t ws_size,
                              hipStream_t stream) {
  (void)in_sizes; (void)n_in; (void)out_size; (void)d_ws; (void)ws_size;
  const float* x = (const float*)d_in[0];
  float* xl = (float*)d_out;
  float* xh = xl + (size_t)64 * T_LEN * N_LEN;   // outputs concatenated (xl, xh)

  dim3 grid(N_LEN / NT, 64);                 // 32 n-tiles x 64 batches
  const size_t shmem = (size_t)(T_LEN + 520 + A2_ROWS) * PITCH * sizeof(float);
  dwt2_lowhigh_kernel<<<grid, 256, shmem, stream>>>(x, xl, xh);
}